// STAGATE2_16372415332908
// MI455X (gfx1250) — compile-verified
//
#include <hip/hip_runtime.h>
#include <math.h>

typedef __attribute__((ext_vector_type(2))) float v2f;
typedef __attribute__((ext_vector_type(8))) float v8f;

#define Nn  30000
#define Ee  180000
#define INF_ 3000
#define Hh  512
#define OUTc 30

// ---------- float <-> order-preserving unsigned key ----------
__device__ __forceinline__ unsigned f2key(float f) {
    unsigned u = __float_as_uint(f);
    return (u & 0x80000000u) ? ~u : (u | 0x80000000u);
}
__device__ __forceinline__ float key2f(unsigned k) {
    unsigned u = (k & 0x80000000u) ? (k & 0x7fffffffu) : ~k;
    return __uint_as_float(u);
}

#define WMMA_F32(a, b, c) \
    __builtin_amdgcn_wmma_f32_16x16x4_f32(false, (a), false, (b), (short)0, (c), false, false)

// ---------- fp32 WMMA GEMM: C[M,NCOLS] = A[M,K] * B ----------
// BT=true : B stored [NCOLS,K] row-major (C = A * B^T)
// BT=false: B stored [K,NCOLS] row-major (C = A * B)
// One wave per 16x64 output tile; K multiple of 4; M multiple of 16.
// Out-of-range columns: index CLAMPED for loads (results discarded at store).
// Software-pipelined: loads for k+4 issued before WMMAs consuming stage k.
template<bool BT, int KK, int NCOLS>
__global__ __launch_bounds__(32)
void gemm_wmma_f32(const float* __restrict__ A, const float* __restrict__ B,
                   float* __restrict__ C) {
    const int lane = threadIdx.x & 31;
    const int half = lane >> 4;        // 0: K pair {0,1}, 1: K pair {2,3}
    const int l16  = lane & 15;
    const int tm = blockIdx.x * 16;
    const int tn = blockIdx.y * 64;

    v8f c0 = {}, c1 = {}, c2 = {}, c3 = {};

    const float* ap = A + (size_t)(tm + l16) * KK + 2 * half;
    const int n0 = tn + l16, n1 = n0 + 16, n2 = n0 + 32, n3 = n0 + 48;
    const int n0c = n0 < NCOLS ? n0 : NCOLS - 1;
    const int n1c = n1 < NCOLS ? n1 : NCOLS - 1;
    const int n2c = n2 < NCOLS ? n2 : NCOLS - 1;
    const int n3c = n3 < NCOLS ? n3 : NCOLS - 1;

    if (BT) {
        const float* bp0 = B + (size_t)n0c * KK + 2 * half;
        const float* bp1 = B + (size_t)n1c * KK + 2 * half;
        const float* bp2 = B + (size_t)n2c * KK + 2 * half;
        const float* bp3 = B + (size_t)n3c * KK + 2 * half;
        // prologue: stage k = 0
        v2f a  = *(const v2f*)(ap);
        v2f b0 = *(const v2f*)(bp0);
        v2f b1 = *(const v2f*)(bp1);
        v2f b2 = *(const v2f*)(bp2);
        v2f b3 = *(const v2f*)(bp3);
        for (int k0 = 0; k0 < KK - 4; k0 += 4) {
            // issue loads for next stage first
            v2f an  = *(const v2f*)(ap  + k0 + 4);
            v2f b0n = *(const v2f*)(bp0 + k0 + 4);
            v2f b1n = *(const v2f*)(bp1 + k0 + 4);
            v2f b2n = *(const v2f*)(bp2 + k0 + 4);
            v2f b3n = *(const v2f*)(bp3 + k0 + 4);
            // consume current stage
            c0 = WMMA_F32(a, b0, c0);
            c1 = WMMA_F32(a, b1, c1);
            c2 = WMMA_F32(a, b2, c2);
            c3 = WMMA_F32(a, b3, c3);
            a = an; b0 = b0n; b1 = b1n; b2 = b2n; b3 = b3n;
        }
        c0 = WMMA_F32(a, b0, c0);
        c1 = WMMA_F32(a, b1, c1);
        c2 = WMMA_F32(a, b2, c2);
        c3 = WMMA_F32(a, b3, c3);
    } else {
        const float* Bc0 = B + n0c;
        const float* Bc1 = B + n1c;
        const float* Bc2 = B + n2c;
        const float* Bc3 = B + n3c;
        const size_t rr0 = (size_t)(2 * half) * NCOLS;
        const size_t rr1 = rr0 + NCOLS;
        v2f a, b0, b1, b2, b3;
        a = *(const v2f*)(ap);
        b0.x = Bc0[rr0]; b0.y = Bc0[rr1];
        b1.x = Bc1[rr0]; b1.y = Bc1[rr1];
        b2.x = Bc2[rr0]; b2.y = Bc2[rr1];
        b3.x = Bc3[rr0]; b3.y = Bc3[rr1];
        for (int k0 = 0; k0 < KK - 4; k0 += 4) {
            const size_t r0 = (size_t)(k0 + 4 + 2 * half) * NCOLS;
            const size_t r1 = r0 + NCOLS;
            v2f an, b0n, b1n, b2n, b3n;
            an = *(const v2f*)(ap + k0 + 4);
            b0n.x = Bc0[r0]; b0n.y = Bc0[r1];
            b1n.x = Bc1[r0]; b1n.y = Bc1[r1];
            b2n.x = Bc2[r0]; b2n.y = Bc2[r1];
            b3n.x = Bc3[r0]; b3n.y = Bc3[r1];
            c0 = WMMA_F32(a, b0, c0);
            c1 = WMMA_F32(a, b1, c1);
            c2 = WMMA_F32(a, b2, c2);
            c3 = WMMA_F32(a, b3, c3);
            a = an; b0 = b0n; b1 = b1n; b2 = b2n; b3 = b3n;
        }
        c0 = WMMA_F32(a, b0, c0);
        c1 = WMMA_F32(a, b1, c1);
        c2 = WMMA_F32(a, b2, c2);
        c3 = WMMA_F32(a, b3, c3);
    }

#pragma unroll
    for (int v = 0; v < 8; v++) {
        const int row = tm + v + 8 * half;
        float* crow = C + (size_t)row * NCOLS;
        if (n0 < NCOLS) crow[n0] = c0[v];
        if (n1 < NCOLS) crow[n1] = c1[v];
        if (n2 < NCOLS) crow[n2] = c2[v];
        if (n3 < NCOLS) crow[n3] = c3[v];
    }
}

// ---------- vd[k] = sum_h a1d[h] * W1d[h,k] ----------
__global__ void k_vd(const float* __restrict__ W1d, const float* __restrict__ a1d,
                     float* __restrict__ vd) {
    int k = blockIdx.x * blockDim.x + threadIdx.x;
    if (k >= INF_) return;
    float s = 0.f;
    for (int h = 0; h < Hh; h++) s += a1d[h] * W1d[(size_t)h * INF_ + k];
    vd[k] = s;
}

// ---------- per-node dots: s_d = feat . vd ; s_s = xs1 . a1s ----------
__global__ __launch_bounds__(256)
void k_ss_sd(const float* __restrict__ feat, const float* __restrict__ vd,
             const float* __restrict__ xs1, const float* __restrict__ a1s,
             float* __restrict__ ss, float* __restrict__ sd) {
    __shared__ float rs[256], rd[256];
    const int n = blockIdx.x, t = threadIdx.x;
    float ld = 0.f, ls = 0.f;
    const float* fr = feat + (size_t)n * INF_;
    for (int k = t; k < INF_; k += 256) ld += fr[k] * vd[k];
    const float* xr = xs1 + (size_t)n * Hh;
    for (int h = t; h < Hh; h += 256) ls += xr[h] * a1s[h];
    rd[t] = ld; rs[t] = ls;
    __syncthreads();
    for (int s = 128; s > 0; s >>= 1) {
        if (t < s) { rd[t] += rd[t + s]; rs[t] += rs[t + s]; }
        __syncthreads();
    }
    if (t == 0) { sd[n] = rd[0]; ss[n] = rs[0]; }
}

// ---------- node init: mkey = key(-inf), den = 0 ----------
__global__ void k_node_init(unsigned* __restrict__ mkey, float* __restrict__ den) {
    int n = blockIdx.x * blockDim.x + threadIdx.x;
    if (n >= Nn) return;
    mkey[n] = f2key(-INFINITY);
    den[n] = 0.f;
}

// ---------- e = leaky_relu(ss[src]+sd[dst]); segment max ----------
__global__ void k_edge_e(const float* __restrict__ ss, const float* __restrict__ sd,
                         const int* __restrict__ src, const int* __restrict__ dst,
                         float* __restrict__ e, unsigned* __restrict__ mkey) {
    int j = blockIdx.x * blockDim.x + threadIdx.x;
    if (j >= Ee) return;
    float v = ss[src[j]] + sd[dst[j]];
    v = v > 0.f ? v : 0.2f * v;
    e[j] = v;
    atomicMax(&mkey[dst[j]], f2key(v));
}

__global__ void k_decode_m(const unsigned* __restrict__ mkey, float* __restrict__ mflt) {
    int n = blockIdx.x * blockDim.x + threadIdx.x;
    if (n >= Nn) return;
    float f = key2f(mkey[n]);
    mflt[n] = isinf(f) ? 0.f : f;
}

__global__ void k_exp(float* __restrict__ e, const float* __restrict__ mflt,
                      const int* __restrict__ dst, float* __restrict__ den) {
    int j = blockIdx.x * blockDim.x + threadIdx.x;
    if (j >= Ee) return;
    float ex = expf(e[j] - mflt[dst[j]]);
    e[j] = ex;
    atomicAdd(&den[dst[j]], ex);
}

__global__ void k_alpha(float* __restrict__ e, const float* __restrict__ den,
                        const int* __restrict__ dst) {
    int j = blockIdx.x * blockDim.x + threadIdx.x;
    if (j >= Ee) return;
    e[j] = e[j] / den[dst[j]];
}

// ---------- zero / elu ----------
__global__ void k_zero(float* __restrict__ p, size_t n) {
    size_t i = (size_t)blockIdx.x * blockDim.x + threadIdx.x;
    if (i < n) p[i] = 0.f;
}
__global__ void k_elu(float* __restrict__ p, size_t n) {
    size_t i = (size_t)blockIdx.x * blockDim.x + threadIdx.x;
    if (i < n) { float x = p[i]; p[i] = x > 0.f ? x : expf(x) - 1.f; }
}

// ---------- wave-per-edge scatter: out[dst] += alpha * xs[src] ----------
__global__ __launch_bounds__(256)
void k_agg(const float* __restrict__ xs, const float* __restrict__ alpha,
           const int* __restrict__ src, const int* __restrict__ dst,
           float* __restrict__ out) {
    const int gid = blockIdx.x * blockDim.x + threadIdx.x;
    const int wave = gid >> 5, lane = gid & 31;
    if (wave >= Ee) return;
    const int s = src[wave], d = dst[wave];
    const float a = alpha[wave];
    const float* xr = xs + (size_t)s * Hh;
    float* orow = out + (size_t)d * Hh;
    for (int h = lane; h < Hh; h += 32) atomicAdd(&orow[h], a * xr[h]);
}

// ---------- h2[n,o] = h1[n,:] . W2s[o,:] ----------
__global__ __launch_bounds__(256)
void k_h2(const float* __restrict__ h1, const float* __restrict__ W2s,
          float* __restrict__ out) {
    __shared__ float row[Hh];
    const int n = blockIdx.x;
    for (int h = threadIdx.x; h < Hh; h += blockDim.x) row[h] = h1[(size_t)n * Hh + h];
    __syncthreads();
    for (int o = threadIdx.x; o < OUTc; o += blockDim.x) {
        float s = 0.f;
        const float* w = W2s + (size_t)o * Hh;
        for (int h = 0; h < Hh; h++) s += row[h] * w[h];
        out[(size_t)n * OUTc + o] = s;
    }
}

// ---------- xs3[n,h] = sum_o h2[n,o] * W2s[o,h] ----------
__global__ void k_xs3(const float* __restrict__ h2, const float* __restrict__ W2s,
                      float* __restrict__ xs3) {
    size_t idx = (size_t)blockIdx.x * blockDim.x + threadIdx.x;
    if (idx >= (size_t)Nn * Hh) return;
    const int n = (int)(idx / Hh), h = (int)(idx % Hh);
    float s = 0.f;
    const float* hr = h2 + (size_t)n * OUTc;
    for (int o = 0; o < OUTc; o++) s += hr[o] * W2s[(size_t)o * Hh + h];
    xs3[idx] = s;
}

extern "C" void kernel_launch(void* const* d_in, const int* in_sizes, int n_in,
                              void* d_out, int out_size, void* d_ws, size_t ws_size,
                              hipStream_t stream) {
    const float* features = (const float*)d_in[0];
    const int*   ei       = (const int*)d_in[1];
    const float* W1s      = (const float*)d_in[2];
    const float* W1d      = (const float*)d_in[3];
    const float* a1s      = (const float*)d_in[4];
    const float* a1d      = (const float*)d_in[5];
    const float* W2s      = (const float*)d_in[6];
    const int* src = ei;
    const int* dst = ei + Ee;

    float* h2out = (float*)d_out;                     // [N, OUT]
    float* h4out = (float*)d_out + (size_t)Nn * OUTc; // [N, IN]

    const size_t NH = (size_t)Nn * Hh;
    float* bufA   = (float*)d_ws;        // xs1, later xs3
    float* bufB   = bufA + NH;           // h1, later h3
    float* ealpha = bufB + NH;           // e -> ex -> alpha
    float* ss     = ealpha + Ee;
    float* sd     = ss + Nn;
    unsigned* mkey = (unsigned*)(sd + Nn);
    float* mflt   = (float*)(mkey + Nn);
    float* den    = mflt + Nn;
    float* vd     = den + Nn;            // [IN]

    const int TB = 256;
    // vd = a1d @ W1d  (folds away the second big GEMM)
    k_vd<<<(INF_ + TB - 1) / TB, TB, 0, stream>>>(W1d, a1d, vd);
    // xs1 = features @ W1s^T    [N,H]
    gemm_wmma_f32<true, INF_, Hh><<<dim3(Nn / 16, Hh / 64), 32, 0, stream>>>(features, W1s, bufA);
    // per-node attention scalars
    k_ss_sd<<<Nn, 256, 0, stream>>>(features, vd, bufA, a1s, ss, sd);
    // segment softmax over edges
    k_node_init<<<(Nn + TB - 1) / TB, TB, 0, stream>>>(mkey, den);
    k_edge_e<<<(Ee + TB - 1) / TB, TB, 0, stream>>>(ss, sd, src, dst, ealpha, mkey);
    k_decode_m<<<(Nn + TB - 1) / TB, TB, 0, stream>>>(mkey, mflt);
    k_exp<<<(Ee + TB - 1) / TB, TB, 0, stream>>>(ealpha, mflt, dst, den);
    k_alpha<<<(Ee + TB - 1) / TB, TB, 0, stream>>>(ealpha, den, dst);
    // h1 = segsum(alpha * xs1[src]) ; elu
    k_zero<<<(unsigned)((NH + TB - 1) / TB), TB, 0, stream>>>(bufB, NH);
    k_agg<<<(Ee * 32 + TB - 1) / TB, TB, 0, stream>>>(bufA, ealpha, src, dst, bufB);
    k_elu<<<(unsigned)((NH + TB - 1) / TB), TB, 0, stream>>>(bufB, NH);
    // h2 = h1 @ W2s^T  -> output 1
    k_h2<<<Nn, 256, 0, stream>>>(bufB, W2s, h2out);
    // xs3 = h2 @ W2s   [N,H]  (reuse bufA)
    k_xs3<<<(unsigned)((NH + TB - 1) / TB), TB, 0, stream>>>(h2out, W2s, bufA);
    // h3 = segsum(alpha * xs3[src]) ; elu  (reuse bufB)
    k_zero<<<(unsigned)((NH + TB - 1) / TB), TB, 0, stream>>>(bufB, NH);
    k_agg<<<(Ee * 32 + TB - 1) / TB, TB, 0, stream>>>(bufA, ealpha, src, dst, bufB);
    k_elu<<<(unsigned)((NH + TB - 1) / TB), TB, 0, stream>>>(bufB, NH);
    // h4 = h3 @ W1s    [N,IN] -> output 2
    gemm_wmma_f32<false, Hh, INF_><<<dim3(Nn / 16, (INF_ + 63) / 64), 32, 0, stream>>>(bufB, W1s, h4out);
}